// StochasticKeyNet_87084756894258
// MI455X (gfx1250) — compile-verified
//
#include <hip/hip_runtime.h>
#include <hip/hip_bf16.h>

typedef __attribute__((ext_vector_type(2))) float v2f;
typedef __attribute__((ext_vector_type(8))) float v8f;

#define BATCH 128

// One wave (wave32) computes MT x (16 cout x 16 pixel) output tiles via
// V_WMMA_F32_16X16X4_F32 implicit GEMM. The im2col B-fragment gather is loaded
// once per K-step and reused across the MT independent accumulators.
// All loads are unconditional from clamped in-bounds addresses; boundary
// zeroing is a v_cndmask on the loaded value (no EXEC divergence -> WMMA-safe).
template<int CIN, int H, int W, int COUT, int S, int MT>
__global__ __launch_bounds__(32)
void conv_wmma(const float* __restrict__ U, const float* __restrict__ Wt,
               const float* __restrict__ bias, float* __restrict__ wa) {
  constexpr int Ho = H / S, Wo = W / S;
  constexpr int HWo = Ho * Wo, HWi = H * W;
  constexpr int K = CIN * 9;
  constexpr int NIN  = CIN  * HWi + 1;   // input row length (activations + t)
  constexpr int NOUT = COUT * HWo + 1;   // output row length

  const int lane = threadIdx.x;          // one wave per block
  const int tn = blockIdx.x;             // pixel tile
  const int mg = blockIdx.y;             // cout tile-group (MT tiles each)
  const int b  = blockIdx.z;             // batch
  const int l15  = lane & 15;
  const int koff = (lane >> 4) * 2;      // K-slot base for this half-wave

  const int np = tn * 16 + l15;          // output pixel (B-frag column)
  const int oh = np / Wo, ow = np % Wo;

  const float* __restrict__ Urow = U + (size_t)b * NIN;
  const float t = Urow[NIN - 1];

  v8f acc[MT] = {};

  #pragma unroll
  for (int r = 0; r < 9; ++r) {          // kernel tap kh*3+kw (constant-folded)
    const int kh = r / 3, kw = r % 3;
    const int ih = oh * S + kh - 1;
    const int iw = ow * S + kw - 1;
    const bool valid = (ih >= 0) & (ih < H) & (iw >= 0) & (iw < W);
    const int poff = valid ? (ih * W + iw) : 0;      // clamped, always in-bounds
    for (int ci = 0; ci < CIN; ci += 4) {
      const int k0 = ci + koff;
      const int k1 = k0 + 1;
      int k0c = k0, k1c = k1;
      if constexpr ((CIN & 3) != 0) {                 // K remainder (layer 0 only)
        k0c = k0 < CIN ? k0 : CIN - 1;
        k1c = k1 < CIN ? k1 : CIN - 1;
      }
      // Unconditional gathers, then select-to-zero (v_cndmask, no branches)
      const float bx = Urow[k0c * HWi + poff];
      const float by = Urow[k1c * HWi + poff];
      bool v0 = valid, v1 = valid;
      if constexpr ((CIN & 3) != 0) { v0 = v0 && (k0 < CIN); v1 = v1 && (k1 < CIN); }
      v2f bv;
      bv.x = v0 ? bx : 0.0f;
      bv.y = v1 ? by : 0.0f;
      #pragma unroll
      for (int mt = 0; mt < MT; ++mt) {               // reuse B across MT tiles
        int cm = (mg * MT + mt) * 16 + l15;
        if constexpr ((COUT & 15) != 0) cm = cm < COUT ? cm : COUT - 1;  // pad rows: garbage, never stored
        v2f a;
        a.x = Wt[(size_t)cm * K + k0c * 9 + r];
        a.y = Wt[(size_t)cm * K + k1c * 9 + r];
        acc[mt] = __builtin_amdgcn_wmma_f32_16x16x4_f32(false, a, false, bv,
                                                        (short)0, acc[mt], false, false);
      }
    }
  }

  // Epilogue: y += bias[cout]*t ; wa layout = (cout*HWo + pixel), t at the end
  float* __restrict__ warow = wa + (size_t)b * NOUT;
  #pragma unroll
  for (int mt = 0; mt < MT; ++mt) {
    const int mbase = (mg * MT + mt) * 16 + (lane >> 4) * 8;
    #pragma unroll
    for (int rr = 0; rr < 8; ++rr) {
      const int mm = mbase + rr;
      if ((COUT & 15) == 0 || mm < COUT) {
        warow[(size_t)mm * HWo + tn * 16 + l15] = acc[mt][rr] + bias[mm] * t;
      }
    }
  }
  if (tn == 0 && mg == 0 && lane == 0) warow[NOUT - 1] = t;
}

// u_next[b, p[i]] = relu(d[i] * wa[b, p[i]]) / d[i] — same index read & write -> in place.
__global__ void permute_relu_inplace(float* __restrict__ buf, const int* __restrict__ p,
                                     const float* __restrict__ d, int n) {
  const int i = blockIdx.x * blockDim.x + threadIdx.x;
  const int b = blockIdx.y;
  if (i < n) {
    const int pi = p[i];
    const float di = d[i];
    float* q = buf + (size_t)b * n + pi;
    *q = fmaxf(di * (*q), 0.0f) / di;
  }
}

// feat[b,j] = relu(d9[j] * wa[b, p9[j]]), j < 640 (final permutation is kept)
__global__ void final_feat_kernel(const float* __restrict__ wa, const int* __restrict__ p,
                                  const float* __restrict__ d, float* __restrict__ feat) {
  const int j = blockIdx.x * blockDim.x + threadIdx.x;
  const int b = blockIdx.y;
  if (j < 640) {
    feat[(size_t)b * 640 + j] = fmaxf(d[j] * wa[(size_t)b * 641 + p[j]], 0.0f);
  }
}

// out(128x500) = feat(128x640) @ wfc(500x640)^T + bfc.
// N padded 500->512 by clamping the column index (padded cols never stored),
// so the K loop has zero conditionals and the k-pairs fuse into b64 loads.
__global__ __launch_bounds__(32)
void fc_wmma(const float* __restrict__ feat, const float* __restrict__ wfc,
             const float* __restrict__ bfc, float* __restrict__ out) {
  const int lane = threadIdx.x;
  const int tn = blockIdx.x, tm = blockIdx.y;
  const int l15 = lane & 15, koff = (lane >> 4) * 2;
  const int mrow  = tm * 16 + l15;                 // batch row (A)
  const int ncol0 = tn * 16 + l15;                 // output unit (B)
  const int ncol  = ncol0 < 500 ? ncol0 : 499;     // clamp
  const float* __restrict__ arow = feat + (size_t)mrow * 640 + koff;
  const float* __restrict__ brow = wfc  + (size_t)ncol * 640 + koff;
  v8f c = {};
  #pragma unroll 4
  for (int k = 0; k < 640; k += 4) {
    v2f a, bv;
    a.x  = arow[k];  a.y  = arow[k + 1];           // consecutive -> b64
    bv.x = brow[k];  bv.y = brow[k + 1];           // consecutive -> b64
    c = __builtin_amdgcn_wmma_f32_16x16x4_f32(false, a, false, bv,
                                              (short)0, c, false, false);
  }
  const int mbase = tm * 16 + (lane >> 4) * 8;
  const int nn = tn * 16 + l15;
  #pragma unroll
  for (int rr = 0; rr < 8; ++rr) {
    if (nn < 500) out[(size_t)(mbase + rr) * 500 + nn] = c[rr] + bfc[nn];
  }
}

extern "C" void kernel_launch(void* const* d_in, const int* in_sizes, int n_in,
                              void* d_out, int out_size, void* d_ws, size_t ws_size,
                              hipStream_t stream) {
  (void)in_sizes; (void)n_in; (void)out_size; (void)ws_size;
  const float* x = (const float*)d_in[0];
  const float* Wl[9]; const float* Bl[9]; const int* Pl[9]; const float* Dl[9];
  for (int l = 0; l < 9; ++l) {
    Wl[l] = (const float*)d_in[1 + 4 * l];
    Bl[l] = (const float*)d_in[2 + 4 * l];
    Pl[l] = (const int*)  d_in[3 + 4 * l];
    Dl[l] = (const float*)d_in[4 + 4 * l];
  }
  const float* wfc = (const float*)d_in[37];
  const float* bfc = (const float*)d_in[38];
  float* out = (float*)d_out;

  const size_t bufElems = (size_t)BATCH * 98305;   // largest activation row
  float* bufA = (float*)d_ws;
  float* bufB = bufA + bufElems;
  float* feat = bufB + bufElems;

  const dim3 wg(32, 1, 1);
  #define PERM(buf, l, n) \
    permute_relu_inplace<<<dim3(((n) + 255) / 256, BATCH), 256, 0, stream>>>(buf, Pl[l], Dl[l], (n))

  // grid = (pixel tiles, cout tile-groups, batch); MT tuned so waves >= ~2k per layer
  conv_wmma<  3, 32, 32,  96, 1, 6><<<dim3(64, 1, BATCH), wg, 0, stream>>>(x,    Wl[0], Bl[0], bufA);
  PERM(bufA, 0, 98305);
  conv_wmma< 96, 32, 32,  96, 1, 6><<<dim3(64, 1, BATCH), wg, 0, stream>>>(bufA, Wl[1], Bl[1], bufB);
  PERM(bufB, 1, 98305);
  conv_wmma< 96, 32, 32,  96, 2, 3><<<dim3(16, 2, BATCH), wg, 0, stream>>>(bufB, Wl[2], Bl[2], bufA);
  PERM(bufA, 2, 24577);
  conv_wmma< 96, 16, 16, 192, 1, 6><<<dim3(16, 2, BATCH), wg, 0, stream>>>(bufA, Wl[3], Bl[3], bufB);
  PERM(bufB, 3, 49153);
  conv_wmma<192, 16, 16, 192, 1, 6><<<dim3(16, 2, BATCH), wg, 0, stream>>>(bufB, Wl[4], Bl[4], bufA);
  PERM(bufA, 4, 49153);
  conv_wmma<192, 16, 16, 192, 2, 3><<<dim3( 4, 4, BATCH), wg, 0, stream>>>(bufA, Wl[5], Bl[5], bufB);
  PERM(bufB, 5, 12289);
  conv_wmma<192,  8,  8, 192, 1, 3><<<dim3( 4, 4, BATCH), wg, 0, stream>>>(bufB, Wl[6], Bl[6], bufA);
  PERM(bufA, 6, 12289);
  conv_wmma<192,  8,  8, 192, 1, 3><<<dim3( 4, 4, BATCH), wg, 0, stream>>>(bufA, Wl[7], Bl[7], bufB);
  PERM(bufB, 7, 12289);
  conv_wmma<192,  8,  8,  10, 1, 1><<<dim3( 4, 1, BATCH), wg, 0, stream>>>(bufB, Wl[8], Bl[8], bufA);
  final_feat_kernel<<<dim3(3, BATCH), 256, 0, stream>>>(bufA, Pl[8], Dl[8], feat);
  fc_wmma<<<dim3(32, 8), wg, 0, stream>>>(feat, wfc, bfc, out);
  #undef PERM
}